// CTCLoss_49246095016275
// MI455X (gfx1250) — compile-verified
//
#include <hip/hip_runtime.h>
#include <hip/hip_bf16.h>

// Problem constants (fixed by the reference).
#define Tt 1024
#define Bb 64
#define Vv 128
#define Ll 256
#define Ss (2 * Ll + 1)   // 513
#define NEGF (-1e30f)

#define AS3 __attribute__((address_space(3)))
#define AS1 __attribute__((address_space(1)))

typedef int v4i __attribute__((ext_vector_type(4)));

#if defined(__HIP_DEVICE_COMPILE__) && __has_builtin(__builtin_amdgcn_global_load_async_to_lds_b128)
#define HAVE_ASYNC_B128 1
#endif

// One lane's share of a 512-byte async copy: global -> LDS, 16 bytes/lane, 32 lanes.
__device__ __forceinline__ void async_b128(const float* __restrict__ gsrc,
                                           float* __restrict__ ldst, int lane) {
#if defined(HAVE_ASYNC_B128)
  AS1 v4i* gp = (AS1 v4i*)(const_cast<float*>(gsrc) + lane * 4);
  AS3 v4i* lp = (AS3 v4i*)(ldst + lane * 4);
  __builtin_amdgcn_global_load_async_to_lds_b128(gp, lp, /*offset=*/0, /*cpol=*/0);
#else
  float4 v = reinterpret_cast<const float4*>(gsrc)[lane];
  reinterpret_cast<float4*>(ldst)[lane] = v;
#endif
}

__device__ __forceinline__ void wait_async0() {
#if defined(HAVE_ASYNC_B128)
#if __has_builtin(__builtin_amdgcn_s_wait_asynccnt)
  __builtin_amdgcn_s_wait_asynccnt(0);
#else
  asm volatile("s_wait_asynccnt 0x0" ::: "memory");
#endif
#endif
}

// ---------------------------------------------------------------------------
// Kernel 1: logZ[b*T + t] = max_v acts[t,b,v] + log(sum_v exp(acts[t,b,v]-max))
// One wave32 per (t,b) row; float4 per lane; shuffle reductions.
// ---------------------------------------------------------------------------
__global__ __launch_bounds__(256) void ctc_logz_kernel(const float* __restrict__ acts,
                                                       float* __restrict__ logZ) {
  const int gid = blockIdx.x * 256 + threadIdx.x;
  const int row = gid >> 5;   // 0 .. T*B-1  (row = t*B + b)
  const int lane = gid & 31;

  float4 v = reinterpret_cast<const float4*>(acts + (size_t)row * Vv)[lane];
  float m = fmaxf(fmaxf(v.x, v.y), fmaxf(v.z, v.w));
#pragma unroll
  for (int off = 16; off; off >>= 1) m = fmaxf(m, __shfl_xor(m, off, 32));
  float s = expf(v.x - m) + expf(v.y - m) + expf(v.z - m) + expf(v.w - m);
#pragma unroll
  for (int off = 16; off; off >>= 1) s += __shfl_xor(s, off, 32);

  if (lane == 0) {
    const int t = row / Bb;
    const int b = row - t * Bb;
    logZ[b * Tt + t] = m + logf(s);
  }
}

// ---------------------------------------------------------------------------
// Kernel 2: forward alpha recursion. One block per batch element, one thread
// per extended-label position s (S=513, block=544=17 waves).
//  - alpha double-buffered in LDS  -> ONE barrier per timestep
//  - emission rows double-buffered in LDS, prefetched one step ahead with
//    async global->LDS b128 copies
//  - full logZ[b, 0..T) row staged to LDS once at start (async)
// ---------------------------------------------------------------------------
__global__ __launch_bounds__(544) void ctc_alpha_kernel(
    const float* __restrict__ acts, const int* __restrict__ labels,
    const int* __restrict__ act_lens, const int* __restrict__ label_lens,
    const float* __restrict__ logZ, float* __restrict__ costs) {
  const int b = blockIdx.x;
  const int tid = threadIdx.x;
  const int lane = tid & 31;
  const bool loader = (tid < 32);

  __shared__ float alphaS[2][Ss];   // double buffer: one barrier per step
  __shared__ float rows[2][Vv];     // emission-row double buffer
  __shared__ float lzS[Tt];         // per-block logZ row (4 KB)

  const int act_len = act_lens[b];
  const int lab_len = label_lens[b];

  const float* actsB = acts + (size_t)b * Vv;   // row t lives at actsB + t*B*V
  const float* lzG = logZ + (size_t)b * Tt;

  // Kick off all startup async copies from wave 0: rows 0,1 + the logZ row.
  if (loader) {
    async_b128(actsB, &rows[0][0], lane);
    async_b128(actsB + (size_t)Bb * Vv, &rows[1][0], lane);
#pragma unroll
    for (int c = 0; c < Tt / 128; ++c)          // 8 chunks of 128 floats
      async_b128(lzG + c * 128, &lzS[c * 128], lane);
  }

  // Per-thread extended label value and skip-transition permission (registers);
  // overlaps with the async copies above.
  int myExt = 0;          // blank for even s
  bool myAllow = false;   // allow alpha[s-2] path
  if (tid < Ss) {
    if (tid & 1) {
      const int li = tid >> 1;
      myExt = labels[b * Ll + li];
      if (tid >= 3) myAllow = (myExt != labels[b * Ll + li - 1]);
    }
  }

  wait_async0();
  __syncthreads();

  // alpha at t=0 -> buffer 0.
  float a = NEGF;
  {
    const float lz0 = lzS[0];
    if (tid == 0) a = rows[0][0] - lz0;          // ext[0] == blank
    else if (tid == 1) a = rows[0][myExt] - lz0; // ext[1] == lab[0]
  }
  if (tid < Ss) alphaS[0][tid] = a;
  __syncthreads();

  // Iteration t: read alphaS[(t-1)&1], write alphaS[t&1]; emission row in
  // rows[t&1]; prefetch row t+1 into rows[(t+1)&1] (last read at t-1, safe).
  for (int t = 1; t < Tt; ++t) {
    const int cur = t & 1;
    if (loader && (t + 1) < Tt) {
      async_b128(actsB + (size_t)(t + 1) * Bb * Vv, &rows[cur ^ 1][0], lane);
    }

    if (tid < Ss) {
      const float emit = rows[cur][myExt] - lzS[t];
      const float p1 = (tid >= 1) ? alphaS[cur ^ 1][tid - 1] : NEGF;
      const float p2 = myAllow ? alphaS[cur ^ 1][tid - 2] : NEGF;
      const float m = fmaxf(fmaxf(a, p1), p2);
      const float anew =
          m + logf(expf(a - m) + expf(p1 - m) + expf(p2 - m)) + emit;
      if (t < act_len) a = anew;       // freeze exhausted sequences
      alphaS[cur][tid] = a;
    }
    wait_async0();                     // row t+1 landed in LDS
    __syncthreads();                   // publish alpha writes + new row
  }

  if (tid == 0) {
    const float* af = alphaS[(Tt - 1) & 1];
    const int end = 2 * lab_len;       // last blank position
    const float x = af[end];
    const float y = (end >= 1) ? af[end - 1] : NEGF;
    const float m = fmaxf(x, y);
    costs[b] = -(m + logf(expf(x - m) + expf(y - m)));
  }
}

// ---------------------------------------------------------------------------
// Kernel 3: deterministic fixed-order reduction of per-batch costs.
// ---------------------------------------------------------------------------
__global__ void ctc_sum_kernel(const float* __restrict__ costs, float* __restrict__ out) {
  if (threadIdx.x == 0 && blockIdx.x == 0) {
    float s = 0.0f;
    for (int i = 0; i < Bb; ++i) s += costs[i];
    out[0] = s;
  }
}

extern "C" void kernel_launch(void* const* d_in, const int* in_sizes, int n_in,
                              void* d_out, int out_size, void* d_ws, size_t ws_size,
                              hipStream_t stream) {
  const float* acts = (const float*)d_in[0];       // (T, B, V) f32
  const int* labels = (const int*)d_in[1];         // (B*L,) i32
  const int* act_lens = (const int*)d_in[2];       // (B,) i32
  const int* label_lens = (const int*)d_in[3];     // (B,) i32
  float* out = (float*)d_out;                      // scalar f32

  float* logZ = (float*)d_ws;                      // T*B floats, [b][t] layout
  float* costs = logZ + (size_t)Tt * Bb;           // B floats

  // 1) softmax log-normalizers: one wave per (t,b) row.
  ctc_logz_kernel<<<(Tt * Bb) / 8, 256, 0, stream>>>(acts, logZ);
  // 2) forward recursion: one block per batch element.
  ctc_alpha_kernel<<<Bb, 544, 0, stream>>>(acts, labels, act_lens, label_lens, logZ, costs);
  // 3) deterministic final sum.
  ctc_sum_kernel<<<1, 32, 0, stream>>>(costs, out);
}